// ResponsePredictor_54778012893415
// MI455X (gfx1250) — compile-verified
//
#include <hip/hip_runtime.h>
#include <hip/hip_bf16.h>
#include <math.h>

#define NB    64
#define Hh    22
#define Ww    22
#define HW    484          // 22*22 spatial / source positions
#define SDIM  8
#define NMAPS (NB * HW)    // 30976 independent 22x22 maps in the cv stacks

typedef float v2f __attribute__((ext_vector_type(2)));
typedef float v8f __attribute__((ext_vector_type(8)));

// ---------------------------------------------------------------------------
// cv stack pass 1: conv3x3 (1->8) + bias, accumulate per-channel sum/sumsq
// partials (deterministic block tree reduction; no atomics).
// ---------------------------------------------------------------------------
__global__ void cv_conv1_stats(const float* __restrict__ x,
                               const float* __restrict__ w,   // 8*1*3*3
                               const float* __restrict__ b,   // 8
                               float* __restrict__ partial)   // NMAPS*16
{
    __shared__ float sx[HW];
    __shared__ float sred[512];
    __shared__ float sw[72];
    __shared__ float sb[8];
    const int map = blockIdx.x;
    const int tid = threadIdx.x;
    if (tid < 72) sw[tid] = w[tid];
    if (tid < 8)  sb[tid] = b[tid];
    for (int i = tid; i < HW; i += 512) sx[i] = x[(size_t)map * HW + i];
    __syncthreads();

    float ls[16];
#pragma unroll
    for (int i = 0; i < 16; ++i) ls[i] = 0.f;

    if (tid < HW) {
        const int py = tid / Ww, px = tid % Ww;
#pragma unroll
        for (int oc = 0; oc < 8; ++oc) {
            float acc = sb[oc];
#pragma unroll
            for (int ky = 0; ky < 3; ++ky) {
                const int ny = py + ky - 1;
                if ((unsigned)ny >= (unsigned)Hh) continue;
#pragma unroll
                for (int kx = 0; kx < 3; ++kx) {
                    const int nx = px + kx - 1;
                    if ((unsigned)nx >= (unsigned)Ww) continue;
                    acc += sw[oc * 9 + ky * 3 + kx] * sx[ny * Ww + nx];
                }
            }
            ls[oc]     = acc;
            ls[8 + oc] = acc * acc;
        }
    }
    for (int si = 0; si < 16; ++si) {
        sred[tid] = ls[si];
        __syncthreads();
        for (int s = 256; s > 0; s >>= 1) {
            if (tid < s) sred[tid] += sred[tid + s];
            __syncthreads();
        }
        if (tid == 0) partial[(size_t)map * 16 + si] = sred[0];
        __syncthreads();
    }
}

// Fixed-order reduction of per-block partials: one block per statistic.
__global__ void reduce_partials(const float* __restrict__ partial, int nblocks,
                                int nstats, float* __restrict__ sums)
{
    __shared__ float sred[512];
    const int s = blockIdx.x;
    const int tid = threadIdx.x;
    float acc = 0.f;
    for (int i = tid; i < nblocks; i += 512) acc += partial[(size_t)i * nstats + s];
    sred[tid] = acc;
    __syncthreads();
    for (int k = 256; k > 0; k >>= 1) {
        if (tid < k) sred[tid] += sred[tid + k];
        __syncthreads();
    }
    if (tid == 0) sums[s] = sred[0];
}

// Fold BN stats + gamma/beta into per-channel mul/add:  y = x*mul + add
__global__ void finalize_bn(const float* __restrict__ sums, const float* __restrict__ g,
                            const float* __restrict__ be, int C, float count,
                            float* __restrict__ mul, float* __restrict__ add)
{
    const int c = threadIdx.x;
    if (c < C) {
        const float mean = sums[c] / count;
        const float var  = sums[C + c] / count - mean * mean;
        const float m    = g[c] * rsqrtf(var + 1e-5f);
        mul[c] = m;
        add[c] = be[c] - mean * m;
    }
}

// ---------------------------------------------------------------------------
// cv stack pass 2: conv1+BN1+ReLU (8ch in LDS) -> conv2 (8->1) + bias.
// Writes raw z (pre-BN2) and accumulates sum/sumsq partials for BN2.
// ---------------------------------------------------------------------------
__global__ void cv_chain(const float* __restrict__ x,
                         const float* __restrict__ w1, const float* __restrict__ b1,
                         const float* __restrict__ mul1, const float* __restrict__ add1,
                         const float* __restrict__ w2, const float* __restrict__ b2,
                         float* __restrict__ zout, float* __restrict__ partial2)
{
    __shared__ float sx[HW];
    __shared__ float sh[8 * HW];
    __shared__ float sred[512];
    __shared__ float sw1[72], sw2[72], sb1[8], sm1[8], sa1[8];
    __shared__ float sb2;
    const int map = blockIdx.x;
    const int tid = threadIdx.x;
    if (tid < 72) { sw1[tid] = w1[tid]; sw2[tid] = w2[tid]; }
    if (tid < 8)  { sb1[tid] = b1[tid]; sm1[tid] = mul1[tid]; sa1[tid] = add1[tid]; }
    if (tid == 0) sb2 = b2[0];
    for (int i = tid; i < HW; i += 512) sx[i] = x[(size_t)map * HW + i];
    __syncthreads();

    if (tid < HW) {
        const int py = tid / Ww, px = tid % Ww;
#pragma unroll
        for (int oc = 0; oc < 8; ++oc) {
            float acc = sb1[oc];
#pragma unroll
            for (int ky = 0; ky < 3; ++ky) {
                const int ny = py + ky - 1;
                if ((unsigned)ny >= (unsigned)Hh) continue;
#pragma unroll
                for (int kx = 0; kx < 3; ++kx) {
                    const int nx = px + kx - 1;
                    if ((unsigned)nx >= (unsigned)Ww) continue;
                    acc += sw1[oc * 9 + ky * 3 + kx] * sx[ny * Ww + nx];
                }
            }
            const float h = acc * sm1[oc] + sa1[oc];
            sh[oc * HW + tid] = fmaxf(h, 0.f);
        }
    }
    __syncthreads();

    float ls0 = 0.f, ls1 = 0.f;
    if (tid < HW) {
        const int py = tid / Ww, px = tid % Ww;
        float acc = sb2;
#pragma unroll
        for (int ic = 0; ic < 8; ++ic) {
#pragma unroll
            for (int ky = 0; ky < 3; ++ky) {
                const int ny = py + ky - 1;
                if ((unsigned)ny >= (unsigned)Hh) continue;
#pragma unroll
                for (int kx = 0; kx < 3; ++kx) {
                    const int nx = px + kx - 1;
                    if ((unsigned)nx >= (unsigned)Ww) continue;
                    acc += sw2[ic * 9 + ky * 3 + kx] * sh[ic * HW + ny * Ww + nx];
                }
            }
        }
        zout[(size_t)map * HW + tid] = acc;
        ls0 = acc;
        ls1 = acc * acc;
    }
    sred[tid] = ls0;
    __syncthreads();
    for (int s = 256; s > 0; s >>= 1) { if (tid < s) sred[tid] += sred[tid + s]; __syncthreads(); }
    if (tid == 0) partial2[(size_t)map * 2 + 0] = sred[0];
    __syncthreads();
    sred[tid] = ls1;
    __syncthreads();
    for (int s = 256; s > 0; s >>= 1) { if (tid < s) sred[tid] += sred[tid + s]; __syncthreads(); }
    if (tid == 0) partial2[(size_t)map * 2 + 1] = sred[0];
}

// Spatial softmax per map (applies folded BN2 affine first). In place.
__global__ void softmax_spatial(float* __restrict__ z, const float* __restrict__ mul2,
                                const float* __restrict__ add2)
{
    __shared__ float sred[512];
    __shared__ float sM, sZ;
    const int map = blockIdx.x;
    const int tid = threadIdx.x;
    const float m = mul2[0], a = add2[0];
    float v = 0.f, L = -INFINITY;
    if (tid < HW) { v = z[(size_t)map * HW + tid] * m + a; L = v; }
    sred[tid] = L;
    __syncthreads();
    for (int s = 256; s > 0; s >>= 1) { if (tid < s) sred[tid] = fmaxf(sred[tid], sred[tid + s]); __syncthreads(); }
    if (tid == 0) sM = sred[0];
    __syncthreads();
    const float e = (tid < HW) ? expf(v - sM) : 0.f;
    sred[tid] = e;
    __syncthreads();
    for (int s = 256; s > 0; s >>= 1) { if (tid < s) sred[tid] += sred[tid + s]; __syncthreads(); }
    if (tid == 0) sZ = sred[0];
    __syncthreads();
    if (tid < HW) z[(size_t)map * HW + tid] = e / sZ;
}

// Softmax over the 484 source positions per (n, pixel); conf = max = 1/Z.
__global__ void softmax_over_p(float* __restrict__ L, const float* __restrict__ mul4,
                               const float* __restrict__ add4, float* __restrict__ conf)
{
    const int n = blockIdx.x;
    const int tid = threadIdx.x;
    if (tid >= HW) return;
    const float mm = mul4[0], aa = add4[0];
    const size_t base = (size_t)n * HW * HW;
    float m = -INFINITY, s = 0.f;
    for (int p = 0; p < HW; ++p) {
        const float v = L[base + (size_t)p * HW + tid] * mm + aa;
        const float nm = fmaxf(m, v);
        s = s * expf(m - nm) + expf(v - nm);
        m = nm;
    }
    const float invZ = 1.f / s;
    conf[n * HW + tid] = invZ;   // max of softmax == exp(m-m)/Z == 1/Z
    for (int p = 0; p < HW; ++p) {
        const size_t idx = base + (size_t)p * HW + tid;
        const float v = L[idx] * mm + aa;
        L[idx] = expf(v - m) * invZ;
    }
}

// ---------------------------------------------------------------------------
// prop_h[n,d,hw] = sum_p P[n,p,hw] * state_prev[n,d,p]  via V_WMMA_F32_16X16X4_F32
// One wave per 16x16 D tile: M = hw (31 tiles of 16), N = d (8 valid), K = p (484).
// A (16x4 f32): lane l holds M=l&15, VGPR0 K=2*(l>>4), VGPR1 K=2*(l>>4)+1.
// B (4x16 f32): lane l holds N=l&15, VGPR0 K=2*(l>>4), VGPR1 K=2*(l>>4)+1.
// D (16x16 f32): lane l col N=l&15, VGPR i row M=(l>>4)*8+i.
//
// OOB rows of A / cols of B are handled by ADDRESS CLAMPING, not predication:
// a garbage A row m only corrupts D row m, a garbage B col n only corrupts
// D col n, and those rows/cols are never stored.  This keeps EXEC all-1s
// (required by WMMA) and keeps the inner loop free of exec-mask branches so
// loads can be batched/pipelined by the compiler.
// ---------------------------------------------------------------------------
__global__ void wmma_prop(const float* __restrict__ Pm, const float* __restrict__ sp,
                          float* __restrict__ prop)
{
    const int mt = blockIdx.x;       // 0..30
    const int n  = blockIdx.y;       // 0..63
    const int l  = threadIdx.x;      // 0..31 (one wave)
    const int mrow  = l & 15;
    const int khalf = l >> 4;
    const float* Pn = Pm + (size_t)n * HW * HW;
    const float* sn = sp + (size_t)n * SDIM * HW;
    const int m = mt * 16 + mrow;

    const float* aPtr = Pn + (m < HW ? m : (HW - 1));                 // clamp M
    const float* bPtr = sn + (size_t)(mrow < SDIM ? mrow : 0) * HW;   // clamp N

    v8f c = {0.f, 0.f, 0.f, 0.f, 0.f, 0.f, 0.f, 0.f};
#pragma unroll 4
    for (int k0 = 0; k0 < HW; k0 += 4) {
        const int ka = k0 + 2 * khalf;
        v2f A, B;
        A[0] = aPtr[(size_t)ka * HW];
        A[1] = aPtr[(size_t)(ka + 1) * HW];
        B[0] = bPtr[ka];
        B[1] = bPtr[ka + 1];
        c = __builtin_amdgcn_wmma_f32_16x16x4_f32(false, A, false, B,
                                                  (short)0, c, false, false);
    }
#pragma unroll
    for (int i = 0; i < 8; ++i) {
        const int row = khalf * 8 + i;      // M (hw)
        const int col = mrow;               // N (d)
        const int hw = mt * 16 + row;
        if (col < SDIM && hw < HW)
            prop[((size_t)n * SDIM + col) * HW + hw] = c[i];
    }
}

// Materialize pin = concat([prop_h(8), dimp(1), conf(1)]) as (NB,10,HW).
__global__ void build_pin(const float* __restrict__ prop, const float* __restrict__ dimp,
                          const float* __restrict__ conf, float* __restrict__ pin)
{
    const int idx = blockIdx.x * 256 + threadIdx.x;
    if (idx >= NB * 10 * HW) return;
    const int pix = idx % HW;
    const int c   = (idx / HW) % 10;
    const int n   = idx / (10 * HW);
    float v;
    if (c < 8)       v = prop[((size_t)n * 8 + c) * HW + pix];
    else if (c == 8) v = dimp[n * HW + pix];
    else             v = conf[n * HW + pix];
    pin[idx] = v;
}

// Generic direct 3x3 SAME conv on (NB, IC, 22, 22). act: 0=none 1=relu 2=sigmoid.
__global__ void conv2d_direct(const float* __restrict__ in, const float* __restrict__ w,
                              const float* __restrict__ b, float* __restrict__ out,
                              int IC, int OC, int act)
{
    const int idx = blockIdx.x * 256 + threadIdx.x;
    if (idx >= NB * OC * HW) return;
    const int pix = idx % HW;
    const int oc  = (idx / HW) % OC;
    const int n   = idx / (OC * HW);
    const int py = pix / Ww, px = pix % Ww;
    float acc = b[oc];
    for (int ic = 0; ic < IC; ++ic) {
        const float* ip = in + ((size_t)n * IC + ic) * HW;
        const float* wp = w + ((size_t)oc * IC + ic) * 9;
#pragma unroll
        for (int ky = 0; ky < 3; ++ky) {
            const int ny = py + ky - 1;
            if ((unsigned)ny >= (unsigned)Hh) continue;
#pragma unroll
            for (int kx = 0; kx < 3; ++kx) {
                const int nx = px + kx - 1;
                if ((unsigned)nx >= (unsigned)Ww) continue;
                acc += wp[ky * 3 + kx] * ip[ny * Ww + nx];
            }
        }
    }
    if (act == 1)      acc = fmaxf(acc, 0.f);
    else if (act == 2) acc = 1.f / (1.f + expf(-acc));
    out[((size_t)n * OC + oc) * HW + pix] = acc;
}

// Adaptive max pool (1x1) over dimp and fused per batch.
__global__ void pool_max(const float* __restrict__ dimp, const float* __restrict__ fused,
                         float* __restrict__ pool)
{
    __shared__ float sred[512];
    const int n = blockIdx.x;
    const int tid = threadIdx.x;
    float v = (tid < HW) ? dimp[n * HW + tid] : -INFINITY;
    sred[tid] = v;
    __syncthreads();
    for (int k = 256; k > 0; k >>= 1) { if (tid < k) sred[tid] = fmaxf(sred[tid], sred[tid + k]); __syncthreads(); }
    if (tid == 0) pool[n * 2 + 0] = sred[0];
    __syncthreads();
    v = (tid < HW) ? fused[n * HW + tid] : -INFINITY;
    sred[tid] = v;
    __syncthreads();
    for (int k = 256; k > 0; k >>= 1) { if (tid < k) sred[tid] = fmaxf(sred[tid], sred[tid + k]); __syncthreads(); }
    if (tid == 0) pool[n * 2 + 1] = sred[0];
}

// ConvGRU cell: stacked = [dimp, fused, pool_d, pool_f, prop_h(8)] (12 ch in LDS).
// Writes d_out: channel 0 = fused, channels 1..8 = new state.
__global__ void gru_update(const float* __restrict__ prop, const float* __restrict__ dimp,
                           const float* __restrict__ fused, const float* __restrict__ pool,
                           const float* __restrict__ rw, const float* __restrict__ rb,
                           const float* __restrict__ uw, const float* __restrict__ ub,
                           const float* __restrict__ ow, const float* __restrict__ ob,
                           float* __restrict__ out)
{
    __shared__ float sin_[12 * HW];
    __shared__ float s2[8 * HW];
    const int n = blockIdx.x;
    const int tid = threadIdx.x;
    const float pd = pool[n * 2 + 0], pf = pool[n * 2 + 1];
    for (int i = tid; i < HW; i += 512) {
        sin_[0 * HW + i] = dimp[n * HW + i];
        sin_[1 * HW + i] = fused[n * HW + i];
        sin_[2 * HW + i] = pd;
        sin_[3 * HW + i] = pf;
#pragma unroll
        for (int d = 0; d < 8; ++d)
            sin_[(4 + d) * HW + i] = prop[((size_t)n * 8 + d) * HW + i];
    }
    __syncthreads();

    float u[8];
    if (tid < HW) {
        const int py = tid / Ww, px = tid % Ww;
#pragma unroll
        for (int d = 0; d < 8; ++d) {
            float au = ub[d], ar = rb[d];
            for (int ic = 0; ic < 12; ++ic) {
                const float* sp = &sin_[ic * HW];
                const float* wu = &uw[(d * 12 + ic) * 9];
                const float* wr = &rw[(d * 12 + ic) * 9];
#pragma unroll
                for (int ky = 0; ky < 3; ++ky) {
                    const int ny = py + ky - 1;
                    if ((unsigned)ny >= (unsigned)Hh) continue;
#pragma unroll
                    for (int kx = 0; kx < 3; ++kx) {
                        const int nx = px + kx - 1;
                        if ((unsigned)nx >= (unsigned)Ww) continue;
                        const float v = sp[ny * Ww + nx];
                        au += wu[ky * 3 + kx] * v;
                        ar += wr[ky * 3 + kx] * v;
                    }
                }
            }
            u[d] = 1.f / (1.f + expf(-au));
            const float r = 1.f / (1.f + expf(-ar));
            s2[d * HW + tid] = r * sin_[(4 + d) * HW + tid];
        }
    }
    __syncthreads();

    if (tid < HW) {
        const int py = tid / Ww, px = tid % Ww;
        out[(size_t)n * 9 * HW + tid] = sin_[1 * HW + tid];   // fused -> channel 0
#pragma unroll
        for (int d = 0; d < 8; ++d) {
            float ao = ob[d];
            for (int ic = 0; ic < 12; ++ic) {
                const float* sp = (ic < 4) ? &sin_[ic * HW] : &s2[(ic - 4) * HW];
                const float* wo = &ow[(d * 12 + ic) * 9];
#pragma unroll
                for (int ky = 0; ky < 3; ++ky) {
                    const int ny = py + ky - 1;
                    if ((unsigned)ny >= (unsigned)Hh) continue;
#pragma unroll
                    for (int kx = 0; kx < 3; ++kx) {
                        const int nx = px + kx - 1;
                        if ((unsigned)nx >= (unsigned)Ww) continue;
                        ao += wo[ky * 3 + kx] * sp[ny * Ww + nx];
                    }
                }
            }
            const float o  = tanhf(ao);
            const float pv = sin_[(4 + d) * HW + tid];
            out[(size_t)n * 9 * HW + (1 + d) * HW + tid] = pv * (1.f - u[d]) + o * u[d];
        }
    }
}

extern "C" void kernel_launch(void* const* d_in, const int* in_sizes, int n_in,
                              void* d_out, int out_size, void* d_ws, size_t ws_size,
                              hipStream_t stream)
{
    (void)in_sizes; (void)n_in; (void)out_size; (void)ws_size;
    const float* cost   = (const float*)d_in[0];
    const float* stprev = (const float*)d_in[1];
    const float* dimp   = (const float*)d_in[2];
    const float* cv1_w1 = (const float*)d_in[3];
    const float* cv1_b1 = (const float*)d_in[4];
    const float* cv1_g1 = (const float*)d_in[5];
    const float* cv1_be1= (const float*)d_in[6];
    const float* cv1_w2 = (const float*)d_in[7];
    const float* cv1_b2 = (const float*)d_in[8];
    const float* cv1_g2 = (const float*)d_in[9];
    const float* cv1_be2= (const float*)d_in[10];
    const float* cv2_w1 = (const float*)d_in[11];
    const float* cv2_b1 = (const float*)d_in[12];
    const float* cv2_g1 = (const float*)d_in[13];
    const float* cv2_be1= (const float*)d_in[14];
    const float* cv2_w2 = (const float*)d_in[15];
    const float* cv2_b2 = (const float*)d_in[16];
    const float* cv2_g2 = (const float*)d_in[17];
    const float* cv2_be2= (const float*)d_in[18];
    const float* rp1_w  = (const float*)d_in[19];
    const float* rp1_b  = (const float*)d_in[20];
    const float* rp2_w  = (const float*)d_in[21];
    const float* rp2_b  = (const float*)d_in[22];
    const float* resp_w = (const float*)d_in[23];
    const float* resp_b = (const float*)d_in[24];
    const float* gru_rw = (const float*)d_in[25];
    const float* gru_rb = (const float*)d_in[26];
    const float* gru_uw = (const float*)d_in[27];
    const float* gru_ub = (const float*)d_in[28];
    const float* gru_ow = (const float*)d_in[29];
    const float* gru_ob = (const float*)d_in[30];

    float* ws = (float*)d_ws;
    size_t off = 0;
    float* bufA    = ws + off; off += (size_t)NMAPS * HW;      // 60 MB
    float* bufB    = ws + off; off += (size_t)NMAPS * HW;      // 60 MB
    float* prop    = ws + off; off += (size_t)NB * SDIM * HW;
    float* conf    = ws + off; off += (size_t)NB * HW;
    float* fusedb  = ws + off; off += (size_t)NB * HW;
    float* pool    = ws + off; off += (size_t)NB * 2;
    float* sums    = ws + off; off += 16;
    float* bn      = ws + off; off += 64;
    float* partial = ws + off; off += (size_t)NMAPS * 16;
    // bufA is dead after cv2's cv_chain; alias response buffers into it.
    float* r1  = bufA;                                 // (NB,64,HW)
    float* r2  = bufA + (size_t)NB * 64 * HW;          // (NB,32,HW)
    float* pin = r2   + (size_t)NB * 32 * HW;          // (NB,10,HW)

    const float cnt = (float)NMAPS * (float)HW;        // BN count (exact in fp32)

    // ---- cv1 stack ----
    cv_conv1_stats<<<NMAPS, 512, 0, stream>>>(cost, cv1_w1, cv1_b1, partial);
    reduce_partials<<<16, 512, 0, stream>>>(partial, NMAPS, 16, sums);
    finalize_bn<<<1, 32, 0, stream>>>(sums, cv1_g1, cv1_be1, 8, cnt, bn + 0, bn + 8);
    cv_chain<<<NMAPS, 512, 0, stream>>>(cost, cv1_w1, cv1_b1, bn + 0, bn + 8,
                                        cv1_w2, cv1_b2, bufA, partial);
    reduce_partials<<<2, 512, 0, stream>>>(partial, NMAPS, 2, sums);
    finalize_bn<<<1, 32, 0, stream>>>(sums, cv1_g2, cv1_be2, 1, cnt, bn + 16, bn + 17);
    softmax_spatial<<<NMAPS, 512, 0, stream>>>(bufA, bn + 16, bn + 17);

    // ---- cv2 stack ----
    cv_conv1_stats<<<NMAPS, 512, 0, stream>>>(bufA, cv2_w1, cv2_b1, partial);
    reduce_partials<<<16, 512, 0, stream>>>(partial, NMAPS, 16, sums);
    finalize_bn<<<1, 32, 0, stream>>>(sums, cv2_g1, cv2_be1, 8, cnt, bn + 18, bn + 26);
    cv_chain<<<NMAPS, 512, 0, stream>>>(bufA, cv2_w1, cv2_b1, bn + 18, bn + 26,
                                        cv2_w2, cv2_b2, bufB, partial);
    reduce_partials<<<2, 512, 0, stream>>>(partial, NMAPS, 2, sums);
    finalize_bn<<<1, 32, 0, stream>>>(sums, cv2_g2, cv2_be2, 1, cnt, bn + 34, bn + 35);
    softmax_over_p<<<NB, 512, 0, stream>>>(bufB, bn + 34, bn + 35, conf);

    // ---- state propagation GEMM (WMMA f32 16x16x4) ----
    dim3 gp((HW + 15) / 16, NB);   // 31 x 64 waves
    wmma_prop<<<gp, 32, 0, stream>>>(bufB, stprev, prop);

    // ---- response prediction ----
    build_pin<<<(NB * 10 * HW + 255) / 256, 256, 0, stream>>>(prop, dimp, conf, pin);
    conv2d_direct<<<(NB * 64 * HW + 255) / 256, 256, 0, stream>>>(pin, rp1_w, rp1_b, r1, 10, 64, 1);
    conv2d_direct<<<(NB * 32 * HW + 255) / 256, 256, 0, stream>>>(r1, rp2_w, rp2_b, r2, 64, 32, 1);
    conv2d_direct<<<(NB * 1 * HW + 255) / 256, 256, 0, stream>>>(r2, resp_w, resp_b, fusedb, 32, 1, 2);

    // ---- ConvGRU state update ----
    pool_max<<<NB, 512, 0, stream>>>(dimp, fusedb, pool);
    gru_update<<<NB, 512, 0, stream>>>(prop, dimp, fusedb, pool,
                                       gru_rw, gru_rb, gru_uw, gru_ub, gru_ow, gru_ob,
                                       (float*)d_out);
}